// BareDotProdAttnEncoder_90821378441675
// MI455X (gfx1250) — compile-verified
//
#include <hip/hip_runtime.h>
#include <hip/hip_bf16.h>
#include <stdint.h>

#define BATCH 32
#define NODES 2048
#define DIM   512

typedef __attribute__((ext_vector_type(2))) float v2f;
typedef __attribute__((ext_vector_type(8))) float v8f;

// ---------------------------------------------------------------------------
// Kernel 1: beta[b][i] = exp(<x_i, x_i>) for all nodes, computed on the matrix
// pipe. For a 16-node tile T, D = X_T * X_T^T via V_WMMA_F32_16X16X4_F32
// accumulating over K=512 in steps of 4; the diagonal of D holds the self-dots.
// Per the ISA f32 layouts, A(16x4) and B(4x16)=A^T occupy identical registers,
// so one float2 load per lane per K-step feeds both operands.
// ---------------------------------------------------------------------------
__global__ __launch_bounds__(32)
void beta_kernel(const float* __restrict__ emb, float* __restrict__ beta) {
  const int tile          = blockIdx.x;        // 0 .. BATCH*NODES/16 - 1
  const int tilesPerBatch = NODES / 16;        // 128
  const int b             = tile / tilesPerBatch;
  const int row0          = (tile % tilesPerBatch) * 16;
  const int lane          = threadIdx.x;       // 0..31
  const int m             = lane & 15;         // matrix row this lane carries
  const int khalf         = lane >> 4;         // 0: K={0,1}, 1: K={2,3}

  const float* rowp =
      emb + (size_t)(b * NODES + row0 + m) * DIM + khalf * 2;

  v8f c = {};
#pragma unroll 8
  for (int k0 = 0; k0 < DIM; k0 += 4) {
    v2f a = *(const v2f*)(rowp + k0);
    // D = A*B + C; A == B == X_tile chunk. 8 args: negA,A,negB,B,cmod,C,rA,rB
    c = __builtin_amdgcn_wmma_f32_16x16x4_f32(false, a, false, a,
                                              (short)0, c, false, false);
  }

  // Diagonal extraction (f32 16x16 C/D layout):
  //  lanes 0..15  hold (M=v,   N=lane)    in element v
  //  lanes 16..31 hold (M=v+8, N=lane-16) in element v
  // -> D[l][l] (l<8)  : lane l,    element l
  // -> D[l][l] (l>=8) : lane l+16, element l-8  => lanes 24..31
  int sel = -1, outIdx = -1;
  if (lane < 8)        { sel = lane;      outIdx = row0 + lane; }
  else if (lane >= 24) { sel = lane - 24; outIdx = row0 + (lane - 16); }

  float d = 0.0f;
#pragma unroll
  for (int v = 0; v < 8; ++v)
    if (sel == v) d = c[v];

  if (outIdx >= 0)
    beta[(size_t)b * NODES + outIdx] = __expf(d);
}

// ---------------------------------------------------------------------------
// Kernel 2: the serial scan. One wave per batch => no barriers on the critical
// path (single-wave workgroup barriers are S_NOP), dot-reduce is a 5-level
// shfl_xor butterfly, and x_{i+1} streams into an LDS double buffer via
// GLOBAL_LOAD_ASYNC_TO_LDS_B128 (ASYNCcnt) while step i computes.
// h state lives in d_out (128 MB total -> resident in the 192 MB L2, so the
// data-dependent parent gather is an L2 hit).
// ---------------------------------------------------------------------------
__global__ __launch_bounds__(32)
void scan_kernel(const float* __restrict__ emb,
                 const int*   __restrict__ conn,
                 const float* __restrict__ beta,
                 float*       __restrict__ out) {
  const int b    = blockIdx.x;
  const int lane = threadIdx.x;                // 0..31

  const float* X = emb  + (size_t)b * NODES * DIM;
  float*       H = out  + (size_t)b * NODES * DIM;
  const int*   P = conn + (size_t)b * NODES;
  const float* B = beta + (size_t)b * NODES;

  extern __shared__ float smem[];              // [2][DIM] double buffer

  unsigned ldsBase = 0;
#if __has_builtin(__builtin_amdgcn_groupstaticsize)
  ldsBase = __builtin_amdgcn_groupstaticsize();  // dynamic-LDS base offset
#endif

  // Each lane owns 4 float4 chunks strided so every load instruction covers a
  // contiguous 512 B across the wave: element base e[c] = c*128 + lane*4.
  int e[4];
#pragma unroll
  for (int c = 0; c < 4; ++c) e[c] = c * 128 + lane * 4;

  const uint64_t xbase = (uint64_t)(uintptr_t)X;

  // h_0 = x_0
#pragma unroll
  for (int c = 0; c < 4; ++c)
    *(float4*)(H + e[c]) = *(const float4*)(X + e[c]);

  // Prime the pipeline: async-load x_1 into buffer 0 (4 x b128 per lane).
#pragma unroll
  for (int c = 0; c < 4; ++c) {
    unsigned ldsAddr = ldsBase + (unsigned)(e[c] * sizeof(float));
    unsigned vOff    = (unsigned)((1 * DIM + e[c]) * sizeof(float));
    asm volatile("global_load_async_to_lds_b128 %0, %1, %2"
                 :: "v"(ldsAddr), "v"(vOff), "s"(xbase)
                 : "memory");
  }

  for (int i = 1; i < NODES; ++i) {
    const int cur = (i - 1) & 1;               // buffer holding x_i
    const unsigned curOff = cur ? DIM : 0;
    const unsigned nxtOff = cur ? 0 : DIM;

    if (i + 1 < NODES) {
      // Stream x_{i+1} into the other buffer, then wait only for x_i
      // (<=4 outstanding are the 4 loads just issued).
#pragma unroll
      for (int c = 0; c < 4; ++c) {
        unsigned ldsAddr = ldsBase + (unsigned)((nxtOff + e[c]) * sizeof(float));
        unsigned vOff    = (unsigned)(((i + 1) * DIM + e[c]) * sizeof(float));
        asm volatile("global_load_async_to_lds_b128 %0, %1, %2"
                     :: "v"(ldsAddr), "v"(vOff), "s"(xbase)
                     : "memory");
      }
      asm volatile("s_wait_asynccnt 4" ::: "memory");
    } else {
      asm volatile("s_wait_asynccnt 0" ::: "memory");
    }

    const int p = P[i];                        // uniform -> scalar load

    float4 xv[4], ph[4];
#pragma unroll
    for (int c = 0; c < 4; ++c)
      xv[c] = *(const float4*)(smem + curOff + e[c]);
#pragma unroll
    for (int c = 0; c < 4; ++c)
      ph[c] = *(const float4*)(H + (size_t)p * DIM + e[c]);

    float partial = 0.0f;
#pragma unroll
    for (int c = 0; c < 4; ++c)
      partial += ph[c].x * xv[c].x + ph[c].y * xv[c].y +
                 ph[c].z * xv[c].z + ph[c].w * xv[c].w;

    // Full-wave butterfly reduce; every lane ends with the dot product.
#pragma unroll
    for (int off = 16; off >= 1; off >>= 1)
      partial += __shfl_xor(partial, off, 32);

    const float alpha = __expf(partial);
    const float bet   = B[i];
    const float inv   = 1.0f / (alpha + bet + 1e-15f);
    const float wh    = alpha * inv;
    const float wx    = bet * inv;

#pragma unroll
    for (int c = 0; c < 4; ++c) {
      float4 h;
      h.x = wh * ph[c].x + wx * xv[c].x;
      h.y = wh * ph[c].y + wx * xv[c].y;
      h.z = wh * ph[c].z + wx * xv[c].z;
      h.w = wh * ph[c].w + wx * xv[c].w;
      *(float4*)(H + (size_t)i * DIM + e[c]) = h;
    }
    // Same-wave store->load to global is hardware-ordered; no fence needed.
  }
}

extern "C" void kernel_launch(void* const* d_in, const int* in_sizes, int n_in,
                              void* d_out, int out_size, void* d_ws, size_t ws_size,
                              hipStream_t stream) {
  const float* emb  = (const float*)d_in[0];   // (32, 2048, 512) f32
  const int*   conn = (const int*)  d_in[1];   // (32, 2048) i32
  // d_in[2] = node_mask (all ones; unused, as in the reference)
  float* out  = (float*)d_out;                 // (32, 2048, 512) f32
  float* beta = (float*)d_ws;                  // 32*2048 f32 scratch (256 KB)

  // Parallel pre-pass: beta via WMMA (off the serial critical path).
  beta_kernel<<<dim3(BATCH * NODES / 16), dim3(32), 0, stream>>>(emb, beta);

  // Serial scan: one wave per batch, async-LDS double-buffered x prefetch.
  const size_t smemBytes = 2 * DIM * sizeof(float);
  scan_kernel<<<dim3(BATCH), dim3(32), smemBytes, stream>>>(emb, conn, beta, out);
}